// LowRankRedistributor_54838142435803
// MI455X (gfx1250) — compile-verified
//
#include <hip/hip_runtime.h>

typedef __attribute__((ext_vector_type(2))) float v2f;
typedef __attribute__((ext_vector_type(8))) float v8f;

#define ALPHA_F 0.4f
#define EPS_F   1.1920929e-07f
#define BATCH   256
#define VOCAB   128000
#define RANK    64

#define KCHUNKS 100
#define KCHUNK  (VOCAB / KCHUNKS)   // 1280, multiple of 4

// ---------------------------------------------------------------------------
// ws layout: [0]=amax|L| bits, [1]=amax|Lp| bits, P (256x64 f32) at +256B
// ---------------------------------------------------------------------------
__global__ __launch_bounds__(256) void init_ws_kernel(float* __restrict__ P,
                                                      unsigned* __restrict__ amax) {
  int i = blockIdx.x * blockDim.x + threadIdx.x;
  if (i < BATCH * RANK) P[i] = 0.0f;
  if (i < 2) amax[i] = 0u;
}

// ---------------------------------------------------------------------------
// Phase A: P = L @ W  (split-K, atomicAdd reduction), fused amax(|L|).
// One wave handles one (m-tile, k-chunk), all 4 N-tiles (N=64) so L is read
// exactly once; W re-reads hit L2 (33 MB << 192 MB).
// ---------------------------------------------------------------------------
__global__ __launch_bounds__(256)
void gemm_lw_kernel(const float* __restrict__ L, const float* __restrict__ W,
                    float* __restrict__ P, unsigned* __restrict__ amaxL) {
  const int lane = threadIdx.x & 31;
  const int wave = blockIdx.x * (blockDim.x >> 5) + (threadIdx.x >> 5);
  const int mt = wave & 15;          // 16 M-tiles of 16 rows
  const int kc = wave >> 4;          // K chunk id
  if (kc >= KCHUNKS) return;

  const int m0 = mt * 16;
  const int ln = lane & 15;
  const int hi = lane >> 4;          // half-wave: K offset 0 or 2
  const int k0 = kc * KCHUNK;

  const float* Arow = L + (size_t)(m0 + ln) * VOCAB;

  v8f acc0 = {}, acc1 = {}, acc2 = {}, acc3 = {};
  float mx = 0.0f;

  for (int k = k0; k < k0 + KCHUNK; k += 4) {
    // A fragment: L[m0+ln][k + hi*2 + {0,1}]  (contiguous float2 per lane)
    v2f a = *(const v2f*)(Arow + k + hi * 2);
    mx = fmaxf(mx, fmaxf(fabsf(a.x), fabsf(a.y)));

    // B fragments: W[k + hi*2 + {0,1}][n0 + ln], n0 = 0,16,32,48
    const float* wb = W + (size_t)(k + hi * 2) * RANK + ln;
    v2f b0, b1, b2, b3;
    b0.x = wb[0];  b0.y = wb[RANK +  0];
    b1.x = wb[16]; b1.y = wb[RANK + 16];
    b2.x = wb[32]; b2.y = wb[RANK + 32];
    b3.x = wb[48]; b3.y = wb[RANK + 48];

    acc0 = __builtin_amdgcn_wmma_f32_16x16x4_f32(false, a, false, b0, (short)0, acc0, false, false);
    acc1 = __builtin_amdgcn_wmma_f32_16x16x4_f32(false, a, false, b1, (short)0, acc1, false, false);
    acc2 = __builtin_amdgcn_wmma_f32_16x16x4_f32(false, a, false, b2, (short)0, acc2, false, false);
    acc3 = __builtin_amdgcn_wmma_f32_16x16x4_f32(false, a, false, b3, (short)0, acc3, false, false);
  }

  // C/D layout: VGPR r, lanes 0-15 -> (m0+r, n), lanes 16-31 -> (m0+8+r, n)
  float* crow = P + (size_t)(m0 + hi * 8) * RANK + ln;
#pragma unroll
  for (int r = 0; r < 8; ++r) {
    atomicAdd(crow + (size_t)r * RANK +  0, acc0[r]);
    atomicAdd(crow + (size_t)r * RANK + 16, acc1[r]);
    atomicAdd(crow + (size_t)r * RANK + 32, acc2[r]);
    atomicAdd(crow + (size_t)r * RANK + 48, acc3[r]);
  }

  // wave32 abs-max reduce, then monotone uint atomicMax (|x| >= 0)
#pragma unroll
  for (int off = 16; off > 0; off >>= 1)
    mx = fmaxf(mx, __shfl_xor(mx, off, 32));
  if (lane == 0) atomicMax(amaxL, __float_as_uint(mx));
}

// ---------------------------------------------------------------------------
// Phase B (WRITE=0): amax|P@W^T - L|.  Phase C (WRITE=1): emit lerp output.
// One wave per 16x16 output tile; 16 chained WMMAs over K=RANK=64.
// P (64 KB) and W (33 MB) live in L2; only L / out touch HBM.
// ---------------------------------------------------------------------------
template <int WRITE>
__global__ __launch_bounds__(256)
void mint_pass_kernel(const float* __restrict__ L, const float* __restrict__ W,
                      const float* __restrict__ P, unsigned* __restrict__ amax,
                      float* __restrict__ out) {
  const int lane = threadIdx.x & 31;
  const int wave = blockIdx.x * (blockDim.x >> 5) + (threadIdx.x >> 5);
  const int NVT = VOCAB / 16;        // 8000 vocab tiles
  const int vt = wave % NVT;
  const int mt = wave / NVT;
  if (mt >= BATCH / 16) return;

  const int m0 = mt * 16, v0 = vt * 16;
  const int ln = lane & 15;
  const int hi = lane >> 4;

  const float* Arow = P + (size_t)(m0 + ln) * RANK;   // A[m][k] = P[m0+m][k]
  const float* Brow = W + (size_t)(v0 + ln) * RANK;   // B[k][n] = W[v0+n][k]

  v8f acc = {};
#pragma unroll
  for (int ks = 0; ks < 16; ++ks) {
    v2f a = *(const v2f*)(Arow + ks * 4 + hi * 2);
    v2f b = *(const v2f*)(Brow + ks * 4 + hi * 2);
    acc = __builtin_amdgcn_wmma_f32_16x16x4_f32(false, a, false, b, (short)0, acc, false, false);
  }

  float scale = 1.0f;
  if (WRITE) {
    float aL  = __uint_as_float(amax[0]);
    float aLp = __uint_as_float(amax[1]);
    scale = aL / fmaxf(aLp, EPS_F);
  }

  float mx = 0.0f;
  const size_t base = (size_t)(m0 + hi * 8) * VOCAB + v0 + ln;
#pragma unroll
  for (int r = 0; r < 8; ++r) {
    const size_t idx = base + (size_t)r * VOCAB;
    const float l  = L[idx];
    const float lp = acc[r] - l;                       // remove basis
    if (WRITE) {
      // lerp(minted, L, alpha) = (1-alpha)*minted + alpha*L
      out[idx] = (1.0f - ALPHA_F) * (lp * scale) + ALPHA_F * l;
    } else {
      mx = fmaxf(mx, fabsf(lp));
    }
  }

  if (!WRITE) {
#pragma unroll
    for (int off = 16; off > 0; off >>= 1)
      mx = fmaxf(mx, __shfl_xor(mx, off, 32));
    if (lane == 0) atomicMax(amax + 1, __float_as_uint(mx));
  }
}

// ---------------------------------------------------------------------------
extern "C" void kernel_launch(void* const* d_in, const int* in_sizes, int n_in,
                              void* d_out, int out_size, void* d_ws, size_t ws_size,
                              hipStream_t stream) {
  // setup_inputs order: token_ids (unused by reference), logits, W
  const float* logits = (const float*)d_in[1];
  const float* W      = (const float*)d_in[2];
  float* out          = (float*)d_out;

  unsigned* amax = (unsigned*)d_ws;
  float* P       = (float*)((char*)d_ws + 256);

  // 1) zero P and amax cells (ws is poisoned; re-init every call)
  init_ws_kernel<<<(BATCH * RANK + 255) / 256, 256, 0, stream>>>(P, amax);

  // 2) P = L @ W  (+ amax|L|): 16 m-tiles x KCHUNKS chunks, 8 waves/block
  gemm_lw_kernel<<<(16 * KCHUNKS) / 8, 256, 0, stream>>>(logits, W, P, amax);

  // 3) amax|P@W^T - L|
  const int tiles = (BATCH / 16) * (VOCAB / 16);      // 128000 waves
  mint_pass_kernel<0><<<tiles / 8, 256, 0, stream>>>(logits, W, P, amax, out);

  // 4) out = lerp(scale*(P@W^T - L), L, alpha)
  mint_pass_kernel<1><<<tiles / 8, 256, 0, stream>>>(logits, W, P, amax, out);
}